// QKBmm_74062416052397
// MI455X (gfx1250) — compile-verified
//
#include <hip/hip_runtime.h>

typedef __attribute__((ext_vector_type(2))) float v2f;
typedef __attribute__((ext_vector_type(4))) float v4f;
typedef __attribute__((ext_vector_type(8))) float v8f;

#define B_  2
#define M_  16
#define QT_ 1024
#define KVT_ 4096
#define H_  128
#define TILE 128
#define LDS_STRIDE 132   // 128 + 4 pad: lane stride 132 mod 64 banks = 4 -> conflict-free

__global__ __launch_bounds__(256, 2)
void qk_bmm_wmma_f32(const float* __restrict__ q,
                     const float* __restrict__ k,
                     float* __restrict__ out) {
  extern __shared__ float smem[];
  float* sQ = smem;                      // [128][132]
  float* sK = smem + TILE * LDS_STRIDE;  // [128][132]

  const int slice  = blockIdx.z;          // b*M + m  (0..31)
  const int qBase  = blockIdx.y * TILE;   // row tile in QT
  const int kvBase = blockIdx.x * TILE;   // col tile in KVT

  const float* qS = q + (size_t)slice * QT_ * H_;
  const float* kS = k + (size_t)slice * KVT_ * H_;
  float* outS     = out + (size_t)slice * QT_ * KVT_;

  const int t  = threadIdx.x;
  const int r0 = t >> 5;        // 0..7
  const int c4 = (t & 31) * 4;  // float col within a 128-float row

  // ---- Stage Q tile and K tile into LDS (coalesced b128 loads) ----
#pragma unroll
  for (int r = r0; r < TILE; r += 8) {
    v4f qv = *(const v4f*)(qS + (size_t)(qBase + r) * H_ + c4);
    *(v4f*)(sQ + r * LDS_STRIDE + c4) = qv;
    v4f kv = *(const v4f*)(kS + (size_t)(kvBase + r) * H_ + c4);
    *(v4f*)(sK + r * LDS_STRIDE + c4) = kv;
  }
  __syncthreads();

  // ---- Per-wave 32x64 block of the 128x128 output tile ----
  // 2 M-strips x 4 N-tiles: 6 LDS b64 loads feed 8 WMMAs (vs 9 for 1x8).
  const int lane   = t & 31;
  const int wave   = t >> 5;       // 0..7
  const int waveM  = wave & 3;     // 4 M blocks of 32 rows
  const int waveN  = wave >> 2;    // 2 N blocks of 64 cols
  const int laneLo = lane & 15;
  const int laneHi = lane >> 4;    // 0 or 1

  // A fragment (16x4 f32): lane holds M = lane&15, K = 2*(lane>>4) + v
  const float* aPtr0 = sQ + (waveM * 32 + laneLo) * LDS_STRIDE + 2 * laneHi;
  const float* aPtr1 = aPtr0 + 16 * LDS_STRIDE;
  // B fragment (4x16 f32): lane holds N = lane&15, K = 2*(lane>>4) + v.
  // B = K-matrix transposed, so b[v] = sK[N-row][h].
  const float* bPtr = sK + (waveN * 64 + laneLo) * LDS_STRIDE + 2 * laneHi;

  v8f acc[2][4];
#pragma unroll
  for (int mi = 0; mi < 2; ++mi)
#pragma unroll
    for (int n = 0; n < 4; ++n) acc[mi][n] = (v8f)(0.0f);

#pragma unroll 4
  for (int k4 = 0; k4 < H_; k4 += 4) {
    v2f a0 = *(const v2f*)(aPtr0 + k4);
    v2f a1 = *(const v2f*)(aPtr1 + k4);
#pragma unroll
    for (int n = 0; n < 4; ++n) {
      v2f b = *(const v2f*)(bPtr + n * 16 * LDS_STRIDE + k4);
      // 8 args: (neg_a, A, neg_b, B, c_mod, C, reuse_a, reuse_b)
      acc[0][n] = __builtin_amdgcn_wmma_f32_16x16x4_f32(
          false, a0, false, b, (short)0, acc[0][n], false, false);
      acc[1][n] = __builtin_amdgcn_wmma_f32_16x16x4_f32(
          false, a1, false, b, (short)0, acc[1][n], false, false);
    }
  }

  // ---- Store: C/D layout: VGPR i -> M = i + 8*(lane>>4); N = lane&15 ----
  const int rowBase = qBase + waveM * 32 + 8 * laneHi;
  const int colBase = kvBase + waveN * 64 + laneLo;
#pragma unroll
  for (int mi = 0; mi < 2; ++mi) {
#pragma unroll
    for (int n = 0; n < 4; ++n) {
#pragma unroll
      for (int i = 0; i < 8; ++i) {
        // Output streamed once, never re-read: non-temporal keeps the
        // L2-resident q/k tiles from being evicted by 512MB of scores.
        __builtin_nontemporal_store(
            acc[mi][n][i],
            outS + (size_t)(rowBase + mi * 16 + i) * KVT_ + colBase + n * 16);
      }
    }
  }
}

extern "C" void kernel_launch(void* const* d_in, const int* in_sizes, int n_in,
                              void* d_out, int out_size, void* d_ws, size_t ws_size,
                              hipStream_t stream) {
  const float* q = (const float*)d_in[0];
  const float* k = (const float*)d_in[1];
  float* out     = (float*)d_out;

  dim3 grid(KVT_ / TILE, QT_ / TILE, B_ * M_);  // (32, 8, 32)
  dim3 block(256);
  size_t shmem = 2u * TILE * LDS_STRIDE * sizeof(float);  // ~132 KB of 320 KB LDS

  qk_bmm_wmma_f32<<<grid, block, shmem, stream>>>(q, k, out);
}